// Ir_Consistency_Loss_19653770346929
// MI455X (gfx1250) — compile-verified
//
#include <hip/hip_runtime.h>

typedef float v2f __attribute__((ext_vector_type(2)));
typedef float v8f __attribute__((ext_vector_type(8)));

#define D 128

// Zero the double accumulator in workspace (d_ws is poisoned to 0xAA once).
__global__ void ircl_init(double* __restrict__ ws) {
    if (threadIdx.x == 0 && blockIdx.x == 0) ws[0] = 0.0;
}

// One wave = 16 edges per batch. Dot products and squared-diff norms are
// computed as the diagonal of V_WMMA_F32_16X16X4_F32 accumulations (fp32).
__global__ __launch_bounds__(256) void ircl_main(
        const float* __restrict__ re,
        const float* __restrict__ irh,
        const long long* __restrict__ src,
        const long long* __restrict__ dst,
        double* __restrict__ ws,
        int n_edges) {
    const int lane   = threadIdx.x & 31;
    const int wave   = (int)((blockIdx.x * blockDim.x + threadIdx.x) >> 5);
    const int nwaves = (int)((gridDim.x * blockDim.x) >> 5);
    const int nbatch = (n_edges + 15) >> 4;

    // WMMA 16x16x4 f32 operand layout:
    //   A (16x4): lane L holds row M=L%16; lanes 0-15 K={0,1}, lanes 16-31 K={2,3}
    //   B (4x16): lane L holds col N=L%16 with the mirrored K pairing
    // => every lane loads a contiguous float2 at [row*D + 4*c + kHi].
    const int eSub = lane & 15;            // edge-in-batch this lane serves
    const int kHi  = (lane >= 16) ? 2 : 0; // K sub-offset within the 4-chunk
    const int comp = lane & 7;             // C-vector component of the diagonal
    // Diagonal (e,e): e=0..7 -> VGPR e lane e;  e=8..15 -> VGPR e-8 lane e+16 (lanes 24-31)
    const bool diagLane = (lane < 8) || (lane >= 16 && lane < 24);
    const int  diagOff  = (lane < 8) ? lane : (lane - 8); // edge-in-batch on diag lanes

    float acc = 0.0f;

    for (int b = wave; b < nbatch; b += nwaves) {
        const int ebase = b * 16;
        int e = ebase + eSub;
        if (e >= n_edges) e = n_edges - 1;      // clamp (select, EXEC stays full)

        const long long sR = src[e];
        const long long dR = dst[e];
        const float* __restrict__ us = re  + sR * D;
        const float* __restrict__ vs = re  + dR * D;
        const float* __restrict__ pu = irh + sR * D;
        const float* __restrict__ pv = irh + dR * D;

        v8f cdot = {0.f,0.f,0.f,0.f,0.f,0.f,0.f,0.f};
        v8f csq  = {0.f,0.f,0.f,0.f,0.f,0.f,0.f,0.f};

        #pragma unroll
        for (int c = 0; c < D / 4; ++c) {
            const int off = c * 4 + kHi;
            v2f a  = *(const v2f*)(us + off);
            v2f bb = *(const v2f*)(vs + off);
            v2f du = *(const v2f*)(pu + off);
            v2f dv = *(const v2f*)(pv + off);
            v2f df = du - dv;
            // D = A*B + C, fp32 matrix pipe; diagonal accumulates per-edge dots
            cdot = __builtin_amdgcn_wmma_f32_16x16x4_f32(
                false, a,  false, bb, (short)0, cdot, false, false);
            csq  = __builtin_amdgcn_wmma_f32_16x16x4_f32(
                false, df, false, df, (short)0, csq,  false, false);
        }

        // Pull this lane's diagonal element (constant-index select chain).
        float ddot = 0.0f, dsq = 0.0f;
        #pragma unroll
        for (int r = 0; r < 8; ++r) {
            ddot = (comp == r) ? cdot[r] : ddot;
            dsq  = (comp == r) ? csq[r]  : dsq;
        }

        const bool valid = diagLane && ((ebase + diagOff) < n_edges);
        acc += valid ? (1.0f - ddot) * dsq : 0.0f;
    }

    // Wave32 butterfly reduction, then one f64 atomic per wave.
    #pragma unroll
    for (int o = 16; o > 0; o >>= 1)
        acc += __shfl_xor(acc, o, 32);
    if (lane == 0)
        atomicAdd(ws, (double)acc);
}

__global__ void ircl_finalize(const double* __restrict__ ws,
                              float* __restrict__ out, int n_edges) {
    if (threadIdx.x == 0 && blockIdx.x == 0)
        out[0] = (float)(ws[0] / (double)n_edges);
}

extern "C" void kernel_launch(void* const* d_in, const int* in_sizes, int n_in,
                              void* d_out, int out_size, void* d_ws, size_t ws_size,
                              hipStream_t stream) {
    const float*     re  = (const float*)d_in[0];
    const float*     irh = (const float*)d_in[1];
    const long long* src = (const long long*)d_in[2];
    const long long* dst = (const long long*)d_in[3];
    const int n_edges = in_sizes[2];

    double* ws  = (double*)d_ws;
    float*  out = (float*)d_out;

    ircl_init<<<1, 1, 0, stream>>>(ws);

    const int threads = 256;          // 8 waves/block on wave32
    const int blocks  = 1024;         // 8192 waves; ~12 batches (of 16 edges) each
    ircl_main<<<blocks, threads, 0, stream>>>(re, irh, src, dst, ws, n_edges);

    ircl_finalize<<<1, 1, 0, stream>>>(ws, out, n_edges);
}